// AttentionAggregator2_54494545052269
// MI455X (gfx1250) — compile-verified
//
#include <hip/hip_runtime.h>
#include <cmath>

typedef __attribute__((ext_vector_type(2))) float v2f;
typedef __attribute__((ext_vector_type(8))) float v8f;

#define N_NODES 50000
#define KNEIB   16
#define DIN     128
#define DEDGE   64
#define DNE     192
#define DHID    32
#define NTILES  3125   // 50000 / 16 exactly

// D = A(16x4) * B(4x16) + C, fp32 WMMA (wave32)
__device__ __forceinline__ v8f wmma_f32(v2f a, v2f b, v8f c) {
  return __builtin_amdgcn_wmma_f32_16x16x4_f32(false, a, false, b, (short)0, c,
                                               false, false);
}

__device__ __forceinline__ float elu1(float v) {
  return v > 0.0f ? v : (__expf(v) - 1.0f);
}

// tanh(x) = (e^{2x}-1)/(e^{2x}+1); clamp keeps exp finite, saturates exactly.
__device__ __forceinline__ float tanh_fast(float x) {
  const float xc = fminf(fmaxf(x, -12.0f), 12.0f);
  const float e = __expf(2.0f * xc);
  return (e - 1.0f) * __builtin_amdgcn_rcpf(e + 1.0f);
}

// Stage a K x NCOL row-major weight into B-fragment order:
//   dst[(kk*NT + nt)*64 + lane*2 + j] = W[4kk + 2*(lane>>4) + j][nt*16 + (lane&15)]
// A B-fragment is then ONE ds_load_b64 at (base + lane*8) + (kk*NT+nt)*256.
template <int KS, int NT, int NCOL>
__device__ __forceinline__ void stageFrag(float* dst, const float* __restrict__ src,
                                          int tid) {
  const int total = KS * NT * 64;
  for (int f = tid; f < total; f += 256) {
    const int j = f & 1;
    const int ln = (f >> 1) & 31;
    const int nt = (f >> 6) % NT;
    const int kk = f / (64 * NT);
    const int rr = ln & 15, hh = ln >> 4;
    dst[f] = src[(4 * kk + 2 * hh + j) * NCOL + nt * 16 + rr];
  }
}

__device__ __forceinline__ v2f fragB(const float* base, int slot, int lane) {
  return *(const v2f*)&base[slot * 64 + lane * 2];
}

// ---------------------------------------------------------------------------
// Kernel A: per 16-node tile (1 wave each):
//   x_att = tanh(x @ Wa1a) @ Wa1b     -> workspace [N,32]
//   out[:, 0:128] = elu(x @ Wfcx)
// ---------------------------------------------------------------------------
__global__ __launch_bounds__(256) void kernelA(
    const float* __restrict__ x, const float* __restrict__ Wa1a,
    const float* __restrict__ Wa1b, const float* __restrict__ Wfcx,
    float* __restrict__ xatt, float* __restrict__ out) {
  __shared__ float sWa1a[DIN * DHID];   // 16 KB, fragment order (KS=32, NT=2)
  __shared__ float sWa1b[DHID * DHID];  // 4 KB,  fragment order (KS=8,  NT=2)
  __shared__ float sWfcx[DIN * DIN];    // 64 KB, fragment order (KS=32, NT=8)
  __shared__ float sStage[8][16 * 36];  // per-wave C->A relayout scratch

  const int tid = threadIdx.x;
  stageFrag<32, 2, DHID>(sWa1a, Wa1a, tid);
  stageFrag<8, 2, DHID>(sWa1b, Wa1b, tid);
  stageFrag<32, 8, DIN>(sWfcx, Wfcx, tid);
  __syncthreads();

  const int wave = tid >> 5, lane = tid & 31;
  const int tile = blockIdx.x * 8 + wave;
  if (tile >= NTILES) return;
  const int r = lane & 15, hb = lane >> 4;
  const int nodeBase = tile * 16;

  // A fragments of the x tile (16 x 128): 32 k-steps, kept for both GEMMs
  v2f a[32];
  const float* xrow = x + (size_t)(nodeBase + r) * DIN + 2 * hb;
#pragma unroll
  for (int kk = 0; kk < 32; ++kk) a[kk] = *(const v2f*)(xrow + 4 * kk);

  // t = x @ Wa1a  (16x32)
  v8f t0 = {}, t1 = {};
#pragma unroll
  for (int kk = 0; kk < 32; ++kk) {
    t0 = wmma_f32(a[kk], fragB(sWa1a, kk * 2 + 0, lane), t0);
    t1 = wmma_f32(a[kk], fragB(sWa1a, kk * 2 + 1, lane), t1);
  }
#pragma unroll
  for (int v = 0; v < 8; ++v) { t0[v] = tanh_fast(t0[v]); t1[v] = tanh_fast(t1[v]); }

  // stage tanh(t) C-layout -> row-major so it can be re-read as an A matrix
  // (same-wave LDS ops are in-order; compiler inserts the minimal DScnt waits)
  float* st = sStage[wave];
#pragma unroll
  for (int v = 0; v < 8; ++v) {
    st[(v + 8 * hb) * 36 + r] = t0[v];
    st[(v + 8 * hb) * 36 + 16 + r] = t1[v];
  }

  // x_att = t @ Wa1b  (16x32), K=32 -> 8 k-steps
  v8f xa0 = {}, xa1 = {};
#pragma unroll
  for (int kk = 0; kk < 8; ++kk) {
    const int kc = 4 * kk + 2 * hb;
    v2f at = *(const v2f*)&st[r * 36 + kc];
    xa0 = wmma_f32(at, fragB(sWa1b, kk * 2 + 0, lane), xa0);
    xa1 = wmma_f32(at, fragB(sWa1b, kk * 2 + 1, lane), xa1);
  }
#pragma unroll
  for (int v = 0; v < 8; ++v) {
    xatt[(size_t)(nodeBase + v + 8 * hb) * 32 + r] = xa0[v];
    xatt[(size_t)(nodeBase + v + 8 * hb) * 32 + 16 + r] = xa1[v];
  }

  // fc_x = x @ Wfcx  (16x128): 8 n-tiles x 32 k-steps
  v8f c[8] = {};
#pragma unroll
  for (int kk = 0; kk < 32; ++kk) {
#pragma unroll
    for (int nt = 0; nt < 8; ++nt)
      c[nt] = wmma_f32(a[kk], fragB(sWfcx, kk * 8 + nt, lane), c[nt]);
  }
#pragma unroll
  for (int nt = 0; nt < 8; ++nt)
#pragma unroll
    for (int v = 0; v < 8; ++v)
      out[(size_t)(nodeBase + v + 8 * hb) * 256 + nt * 16 + r] = elu1(c[nt][v]);
}

// ---------------------------------------------------------------------------
// Kernel B: one node per wave; its 16 neighbors are the WMMA M dimension.
//   ne tile staged once in LDS via global_load_async_to_lds_b128, reused for
//   BOTH the attention GEMM and the ws-weighted aggregation.
//   ws = tanh(ne@Wa2a) @ (Wa2b @ x_att)   (Wa2b folded into the score vector)
//   agg = softmax(ws - 1e7*mask)^T @ ne   -> workspace [N,192]
// ---------------------------------------------------------------------------
__global__ __launch_bounds__(256) void kernelB(
    const float* __restrict__ neibs, const float* __restrict__ edge,
    const float* __restrict__ mask, const float* __restrict__ Wa2a,
    const float* __restrict__ Wa2b, const float* __restrict__ xatt,
    float* __restrict__ agg) {
  __shared__ float sWa2a[DNE * DHID];   // 24 KB, fragment order (KS=48, NT=2)
  __shared__ float sWa2bT[DHID * DHID]; // 4 KB, transposed [c][m]
  __shared__ float sNe[8][16 * 196];    // 98 KB, padded row stride 196
  __shared__ float sSc[8][640];         // [0..575] h stage, [576..607] y, [608..623] ws

  const int tid = threadIdx.x;
  stageFrag<48, 2, DHID>(sWa2a, Wa2a, tid);
  for (int i = tid; i < DHID * DHID; i += 256) {
    const int m = i >> 5, cc = i & 31;
    sWa2bT[cc * 32 + m] = Wa2b[i];
  }
  __syncthreads();

  const int wave = tid >> 5, lane = tid & 31;
  const int node = blockIdx.x * 8 + wave;
  if (node >= N_NODES) return;
  const int r = lane & 15, hb = lane >> 4;

  float* ne = sNe[wave];
  // Async DMA of ne = concat(neibs, edge) 16x192 into LDS (ASYNCcnt-tracked).
  {
    const char* nbase = (const char*)neibs + (size_t)node * KNEIB * DIN * 4;
#pragma unroll
    for (int i = 0; i < 16; ++i) {
      const int idx = lane + 32 * i;  // 512 float4
      const int rr = idx >> 5, c4 = idx & 31;
      const uint64_t g = (uint64_t)(uintptr_t)(nbase + (size_t)idx * 16);
      const uint32_t l = (uint32_t)(uintptr_t)&ne[rr * 196 + c4 * 4];
      asm volatile("global_load_async_to_lds_b128 %0, %1, off" ::"v"(l), "v"(g)
                   : "memory");
    }
    const char* ebase = (const char*)edge + (size_t)node * KNEIB * DEDGE * 4;
#pragma unroll
    for (int i = 0; i < 8; ++i) {
      const int idx = lane + 32 * i;  // 256 float4
      const int rr = idx >> 4, c4 = idx & 15;
      const uint64_t g = (uint64_t)(uintptr_t)(ebase + (size_t)idx * 16);
      const uint32_t l = (uint32_t)(uintptr_t)&ne[rr * 196 + 128 + c4 * 4];
      asm volatile("global_load_async_to_lds_b128 %0, %1, off" ::"v"(l), "v"(g)
                   : "memory");
    }
  }
  asm volatile("s_wait_asynccnt 0" ::: "memory");

  // h = tanh(ne @ Wa2a)  (16x32), K=192 -> 48 k-steps
  v8f h0 = {}, h1 = {};
#pragma unroll
  for (int kk = 0; kk < 48; ++kk) {
    const int kc = 4 * kk + 2 * hb;
    v2f av = *(const v2f*)&ne[r * 196 + kc];
    h0 = wmma_f32(av, fragB(sWa2a, kk * 2 + 0, lane), h0);
    h1 = wmma_f32(av, fragB(sWa2a, kk * 2 + 1, lane), h1);
  }
#pragma unroll
  for (int v = 0; v < 8; ++v) { h0[v] = tanh_fast(h0[v]); h1[v] = tanh_fast(h1[v]); }

  // y = Wa2b(row m) . x_att  (lane m computes y[m]); 32 FMAs per lane.
  const float xc = xatt[(size_t)node * 32 + lane];
  float y = 0.0f;
#pragma unroll
  for (int c2 = 0; c2 < 32; ++c2)
    y = fmaf(sWa2bT[c2 * 32 + lane], __shfl(xc, c2, 32), y);

  float* sc = sSc[wave];
#pragma unroll
  for (int v = 0; v < 8; ++v) {  // stage h C-layout -> row major
    sc[(v + 8 * hb) * 36 + r] = h0[v];
    sc[(v + 8 * hb) * 36 + 16 + r] = h1[v];
  }
  sc[576 + lane] = y;

  // ws = h @ Y  where Y's 16 columns all equal y  -> every D column = scores
  v8f cw = {};
#pragma unroll
  for (int kk = 0; kk < 8; ++kk) {
    const int kc = 4 * kk + 2 * hb;
    v2f av = *(const v2f*)&sc[r * 36 + kc];
    v2f bv = *(const v2f*)&sc[576 + kc];
    cw = wmma_f32(av, bv, cw);
  }

  // masked softmax over 16 neighbors; each lane holds rows v + 8*hb
  const float4* mrow = (const float4*)(mask + (size_t)node * KNEIB + 8 * hb);
  const float4 m0 = mrow[0], m1 = mrow[1];
  float w[8];
  w[0] = cw[0] - 9999999.0f * m0.x;
  w[1] = cw[1] - 9999999.0f * m0.y;
  w[2] = cw[2] - 9999999.0f * m0.z;
  w[3] = cw[3] - 9999999.0f * m0.w;
  w[4] = cw[4] - 9999999.0f * m1.x;
  w[5] = cw[5] - 9999999.0f * m1.y;
  w[6] = cw[6] - 9999999.0f * m1.z;
  w[7] = cw[7] - 9999999.0f * m1.w;
  float mx = -3.0e38f;
#pragma unroll
  for (int v = 0; v < 8; ++v) mx = fmaxf(mx, w[v]);
  mx = fmaxf(mx, __shfl_xor(mx, 16, 32));
  float s = 0.0f;
#pragma unroll
  for (int v = 0; v < 8; ++v) { w[v] = __expf(w[v] - mx); s += w[v]; }
  s += __shfl_xor(s, 16, 32);
  const float inv = 1.0f / s;

  if (r == 0) {
#pragma unroll
    for (int v = 0; v < 8; ++v) sc[608 + v + 8 * hb] = w[v] * inv;
  }

  // agg = ws^T @ ne via WMMA: A rows replicated with ws, B-frags from LDS ne.
  v2f aw[4];
#pragma unroll
  for (int kk = 0; kk < 4; ++kk) {
    const int kc = 4 * kk + 2 * hb;
    aw[kk] = *(const v2f*)&sc[608 + kc];
  }
#pragma unroll
  for (int nt = 0; nt < 12; ++nt) {
    v8f cg = {};
#pragma unroll
    for (int kk = 0; kk < 4; ++kk) {
      const int kc = 4 * kk + 2 * hb;
      v2f bv = {ne[kc * 196 + nt * 16 + r], ne[(kc + 1) * 196 + nt * 16 + r]};
      cg = wmma_f32(aw[kk], bv, cg);
    }
    if (lane < 16) agg[(size_t)node * DNE + nt * 16 + lane] = cg[0];  // row 0
  }
}

// ---------------------------------------------------------------------------
// Kernel C: out[:, 128:256] = elu(agg @ Wfcn), 16-node tiles, Wfcn in LDS.
// ---------------------------------------------------------------------------
__global__ __launch_bounds__(256) void kernelC(
    const float* __restrict__ agg, const float* __restrict__ Wfcn,
    float* __restrict__ out) {
  __shared__ float sW[DNE * DIN];  // 96 KB, fragment order (KS=48, NT=8)
  const int tid = threadIdx.x;
  stageFrag<48, 8, DIN>(sW, Wfcn, tid);
  __syncthreads();

  const int wave = tid >> 5, lane = tid & 31;
  const int tile = blockIdx.x * 8 + wave;
  if (tile >= NTILES) return;
  const int r = lane & 15, hb = lane >> 4;
  const int nodeBase = tile * 16;

  const float* arow = agg + (size_t)(nodeBase + r) * DNE + 2 * hb;
  v8f c[8] = {};
#pragma unroll
  for (int kk = 0; kk < 48; ++kk) {
    v2f av = *(const v2f*)(arow + 4 * kk);
#pragma unroll
    for (int nt = 0; nt < 8; ++nt)
      c[nt] = wmma_f32(av, fragB(sW, kk * 8 + nt, lane), c[nt]);
  }
#pragma unroll
  for (int nt = 0; nt < 8; ++nt)
#pragma unroll
    for (int v = 0; v < 8; ++v)
      out[(size_t)(nodeBase + v + 8 * hb) * 256 + 128 + nt * 16 + r] =
          elu1(c[nt][v]);
}

// ---------------------------------------------------------------------------
extern "C" void kernel_launch(void* const* d_in, const int* in_sizes, int n_in,
                              void* d_out, int out_size, void* d_ws,
                              size_t ws_size, hipStream_t stream) {
  (void)in_sizes; (void)n_in; (void)out_size; (void)ws_size;
  const float* x     = (const float*)d_in[0];
  const float* neibs = (const float*)d_in[1];
  const float* edge  = (const float*)d_in[2];
  const float* mask  = (const float*)d_in[3];
  const float* Wa1a  = (const float*)d_in[4];
  const float* Wa1b  = (const float*)d_in[5];
  const float* Wa2a  = (const float*)d_in[6];
  const float* Wa2b  = (const float*)d_in[7];
  const float* Wfcx  = (const float*)d_in[8];
  const float* Wfcn  = (const float*)d_in[9];
  float* out  = (float*)d_out;
  float* xatt = (float*)d_ws;                    // N*32 floats (6.4 MB)
  float* agg  = xatt + (size_t)N_NODES * 32;     // N*192 floats (38.4 MB)

  const int blocksTile = (NTILES + 7) / 8;       // 391
  const int blocksNode = (N_NODES + 7) / 8;      // 6250
  kernelA<<<blocksTile, 256, 0, stream>>>(x, Wa1a, Wa1b, Wfcx, xatt, out);
  kernelB<<<blocksNode, 256, 0, stream>>>(neibs, edge, mask, Wa2a, Wa2b, xatt, agg);
  kernelC<<<blocksTile, 256, 0, stream>>>(agg, Wfcn, out);
}